// ContextAttnDecoder_48979807043690
// MI455X (gfx1250) — compile-verified
//
#include <hip/hip_runtime.h>
#include <math.h>

// Problem constants (match reference)
#define BB 128
#define TT 1024
#define TCC 1024
#define LL 4
#define VV 50000
#define VTT 50000
#define EE 256
#define HH 512
#define MAX_OOV 50
#define WIDE (VTT + MAX_OOV)   // 50050

typedef __attribute__((ext_vector_type(2))) float v2f;
typedef __attribute__((ext_vector_type(8))) float v8f;

__device__ __forceinline__ float sigmoidf_(float x) {
    return 1.0f / (1.0f + __expf(-x));
}

// ---------------------------------------------------------------------------
// 1. Embedding: x[b,e] = tanh( sum_l emb[ids[b,0,l], e] / L )
// ---------------------------------------------------------------------------
__global__ void embed_kernel(const int* __restrict__ ids,
                             const float* __restrict__ emb,
                             float* __restrict__ x) {
    int b = blockIdx.x;
    int tok0 = ids[b * LL + 0];
    int tok1 = ids[b * LL + 1];
    int tok2 = ids[b * LL + 2];
    int tok3 = ids[b * LL + 3];
    for (int e = threadIdx.x; e < EE; e += blockDim.x) {
        float s = emb[(size_t)tok0 * EE + e] + emb[(size_t)tok1 * EE + e] +
                  emb[(size_t)tok2 * EE + e] + emb[(size_t)tok3 * EE + e];
        x[(size_t)b * EE + e] = tanhf(s * (1.0f / (float)LL));
    }
}

// ---------------------------------------------------------------------------
// 2. Generic GEMM via fp32 WMMA:  out[M x N] (+)= X[M x K] @ W[N x K]^T + bias
//    One wave (32 threads) per 16x16 output tile, K-loop in steps of 4 using
//    V_WMMA_F32_16X16X4_F32.
//    A fragment: lane = half*16 + (m%16); float2 = X[m][k+2*half .. +1]
//    B fragment: lane = half*16 + (n%16); float2 = W[n][k+2*half .. +1]
//    D: vgpr r, lane l -> row tm + r + 8*(l>=16), col tn + (l%16)
// ---------------------------------------------------------------------------
__global__ void gemm_xwT_wmma(const float* __restrict__ X, int ldx,
                              const float* __restrict__ W, int ldw,
                              const float* __restrict__ bias,
                              float* __restrict__ out, int ldout,
                              int K, int accum) {
    int lane = threadIdx.x;        // 0..31, wave32
    int half = lane >> 4;          // 0 or 1
    int l16  = lane & 15;
    int tm = blockIdx.y * 16;
    int tn = blockIdx.x * 16;

    const float* xrow = X + (size_t)(tm + l16) * ldx;
    const float* wrow = W + (size_t)(tn + l16) * ldw;

    v8f c = {};
    for (int k = 0; k < K; k += 4) {
        int kk = k + half * 2;
        v2f a, b;
        a.x = xrow[kk];
        a.y = xrow[kk + 1];
        b.x = wrow[kk];
        b.y = wrow[kk + 1];
        // (neg_a, A, neg_b, B, c_mod, C, reuse_a, reuse_b)
        c = __builtin_amdgcn_wmma_f32_16x16x4_f32(false, a, false, b,
                                                  (short)0, c, false, false);
    }

    int no = tn + l16;
    float bv = bias ? bias[no] : 0.0f;
#pragma unroll
    for (int r = 0; r < 8; ++r) {
        int mo = tm + r + 8 * half;
        float v = c[r] + bv;
        float* p = out + (size_t)mo * ldout + no;
        if (accum) *p += v; else *p = v;
    }
}

// ---------------------------------------------------------------------------
// 3. LSTM pointwise: gates (B x 4H, order i,f,g,o) -> h_new, c_new
//    Written both to scratch (dec) and to the output tail.
// ---------------------------------------------------------------------------
__global__ void lstm_pointwise(const float* __restrict__ gates,
                               const float* __restrict__ c0,
                               float* __restrict__ h_new,
                               float* __restrict__ c_new,
                               float* __restrict__ h_out,
                               float* __restrict__ c_out) {
    int idx = blockIdx.x * blockDim.x + threadIdx.x;
    if (idx >= BB * HH) return;
    int b = idx / HH, h = idx % HH;
    const float* g = gates + (size_t)b * 4 * HH;
    float gi = sigmoidf_(g[h]);
    float gf = sigmoidf_(g[HH + h]);
    float gg = tanhf(g[2 * HH + h]);
    float go = sigmoidf_(g[3 * HH + h]);
    float c = gf * c0[idx] + gi * gg;
    float hn = go * tanhf(c);
    h_new[idx] = hn;
    c_new[idx] = c;
    h_out[idx] = hn;
    c_out[idx] = c;
}

// ---------------------------------------------------------------------------
// 4. Attention scores: scores[b,t] = dot(enc[b,t,:], d[b,:])
//    grid (T/8, B), block 256 = 8 waves; one wave per t, lane-strided dot.
// ---------------------------------------------------------------------------
__global__ void attn_scores_kernel(const float* __restrict__ enc,
                                   const float* __restrict__ d,
                                   float* __restrict__ scores, int Tdim) {
    __shared__ float ds[HH];
    int b = blockIdx.y;
    for (int i = threadIdx.x; i < HH; i += blockDim.x)
        ds[i] = d[(size_t)b * HH + i];
    __syncthreads();

    int wave = threadIdx.x >> 5;
    int lane = threadIdx.x & 31;
    int t = blockIdx.x * 8 + wave;
    const float* row = enc + ((size_t)b * Tdim + t) * HH;
    float acc = 0.0f;
#pragma unroll 4
    for (int j = lane; j < HH; j += 32) acc += row[j] * ds[j];
#pragma unroll
    for (int off = 16; off > 0; off >>= 1) acc += __shfl_down(acc, off, 32);
    if (lane == 0) scores[(size_t)b * Tdim + t] = acc;
}

// ---------------------------------------------------------------------------
// 5. Row softmax in place (n <= a few K), one block per row.
// ---------------------------------------------------------------------------
__global__ void softmax_rows(float* __restrict__ s, int n) {
    float* row = s + (size_t)blockIdx.x * n;
    __shared__ float red[256];
    int tid = threadIdx.x;

    float m = -INFINITY;
    for (int i = tid; i < n; i += blockDim.x) m = fmaxf(m, row[i]);
    red[tid] = m; __syncthreads();
    for (int off = 128; off > 0; off >>= 1) {
        if (tid < off) red[tid] = fmaxf(red[tid], red[tid + off]);
        __syncthreads();
    }
    m = red[0]; __syncthreads();

    float sum = 0.0f;
    for (int i = tid; i < n; i += blockDim.x) {
        float e = __expf(row[i] - m);
        row[i] = e;
        sum += e;
    }
    red[tid] = sum; __syncthreads();
    for (int off = 128; off > 0; off >>= 1) {
        if (tid < off) red[tid] += red[tid + off];
        __syncthreads();
    }
    float inv = 1.0f / red[0];
    for (int i = tid; i < n; i += blockDim.x) row[i] *= inv;
}

// ---------------------------------------------------------------------------
// 6. Attention context: ctx[b,h] = sum_t s[b,t] * enc[b,t,h]
//    One block of H threads per batch row; coalesced over h.
// ---------------------------------------------------------------------------
__global__ void attn_context_kernel(const float* __restrict__ enc,
                                    const float* __restrict__ s,
                                    float* __restrict__ ctx, int Tdim) {
    int b = blockIdx.x;
    int h = threadIdx.x;                       // blockDim == HH
    const float* e = enc + (size_t)b * Tdim * HH + h;
    const float* sr = s + (size_t)b * Tdim;
    float acc = 0.0f;
    for (int t = 0; t < Tdim; ++t) acc += sr[t] * e[(size_t)t * HH];
    ctx[(size_t)b * HH + h] = acc;
}

// ---------------------------------------------------------------------------
// 7. p_gen: sigmoid( [ctx|cctx|dec|x] . gen_W + gen_b + sig_b ),
//    forced to 1.0 when count(context_input[b,:] > 0) == 0.
// ---------------------------------------------------------------------------
__global__ void pgen_kernel(const float* __restrict__ ctx,
                            const float* __restrict__ cctx,
                            const float* __restrict__ dec,
                            const float* __restrict__ x,
                            const float* __restrict__ gen_W,
                            const float* __restrict__ gen_b,
                            const float* __restrict__ sig_b,
                            const int* __restrict__ context_input,
                            float* __restrict__ p_gen) {
    int b = blockIdx.x;
    int tid = threadIdx.x;
    __shared__ float redf[256];
    __shared__ int redi[256];

    const int FEAT = 3 * HH + EE;
    float acc = 0.0f;
    for (int i = tid; i < FEAT; i += blockDim.x) {
        float f;
        if (i < HH)          f = ctx [(size_t)b * HH + i];
        else if (i < 2 * HH) f = cctx[(size_t)b * HH + i - HH];
        else if (i < 3 * HH) f = dec [(size_t)b * HH + i - 2 * HH];
        else                 f = x   [(size_t)b * EE + i - 3 * HH];
        acc += f * gen_W[i];
    }
    int cnt = 0;
    for (int i = tid; i < TCC; i += blockDim.x)
        cnt += (context_input[(size_t)b * TCC + i] > 0) ? 1 : 0;

    redf[tid] = acc; redi[tid] = cnt; __syncthreads();
    for (int off = 128; off > 0; off >>= 1) {
        if (tid < off) { redf[tid] += redf[tid + off]; redi[tid] += redi[tid + off]; }
        __syncthreads();
    }
    if (tid == 0) {
        float pg = sigmoidf_(redf[0] + gen_b[0] + sig_b[0]);
        if (redi[0] == 0) pg = 1.0f;
        p_gen[b] = pg;
    }
}

// ---------------------------------------------------------------------------
// 8. Vocab softmax (online max/sum) scaled by p_gen; zero the OOV columns.
//    Row width WIDE, softmax over first VT columns. One block per row.
// ---------------------------------------------------------------------------
__global__ void vocab_softmax_scale(float* __restrict__ prob,
                                    const float* __restrict__ p_gen) {
    int b = blockIdx.x;
    int tid = threadIdx.x;
    float* row = prob + (size_t)b * WIDE;
    __shared__ float sm[256], ss[256];

    float m = -INFINITY, s = 0.0f;
    for (int i = tid; i < VTT; i += blockDim.x) {
        float v = row[i];
        float mn = fmaxf(m, v);
        s = s * __expf(m - mn) + __expf(v - mn);
        m = mn;
    }
    sm[tid] = m; ss[tid] = s; __syncthreads();
    for (int off = 128; off > 0; off >>= 1) {
        if (tid < off) {
            float m2 = sm[tid + off], s2 = ss[tid + off];
            float mn = fmaxf(sm[tid], m2);
            ss[tid] = ss[tid] * __expf(sm[tid] - mn) + s2 * __expf(m2 - mn);
            sm[tid] = mn;
        }
        __syncthreads();
    }
    m = sm[0];
    float scale = p_gen[b] / ss[0];
    for (int i = tid; i < VTT; i += blockDim.x)
        row[i] = __expf(row[i] - m) * scale;
    for (int i = VTT + tid; i < WIDE; i += blockDim.x)
        row[i] = 0.0f;
}

// ---------------------------------------------------------------------------
// 9. Copy-distribution scatter: prob[b, idx] += (1-p_gen[b]) * cscores[b,t]
// ---------------------------------------------------------------------------
__global__ void scatter_copy_kernel(const int* __restrict__ context_input,
                                    const float* __restrict__ cscores,
                                    const float* __restrict__ p_gen,
                                    float* __restrict__ prob) {
    int idx = blockIdx.x * blockDim.x + threadIdx.x;
    if (idx >= BB * TCC) return;
    int b = idx / TCC;
    float w = 1.0f - p_gen[b];
    int col = context_input[idx];
    atomicAdd(prob + (size_t)b * WIDE + col, w * cscores[idx]);
}

// ---------------------------------------------------------------------------
// 10. log(clip(prob, 1e-10)) -> output
// ---------------------------------------------------------------------------
__global__ void log_kernel(const float* __restrict__ prob,
                           float* __restrict__ out, int n) {
    int i = blockIdx.x * blockDim.x + threadIdx.x;
    if (i < n) out[i] = __logf(fmaxf(prob[i], 1e-10f));
}

// ---------------------------------------------------------------------------
extern "C" void kernel_launch(void* const* d_in, const int* in_sizes, int n_in,
                              void* d_out, int out_size, void* d_ws, size_t ws_size,
                              hipStream_t stream) {
    (void)in_sizes; (void)n_in; (void)out_size; (void)ws_size;

    const int*   input_ids = (const int*)  d_in[0];
    const float* h0        = (const float*)d_in[1];
    const float* c0        = (const float*)d_in[2];
    const float* enc       = (const float*)d_in[3];
    const float* cenc      = (const float*)d_in[4];
    const int*   ctx_in    = (const int*)  d_in[5];
    const float* emb       = (const float*)d_in[6];
    const float* W_ih      = (const float*)d_in[7];
    const float* W_hh      = (const float*)d_in[8];
    const float* b_ih      = (const float*)d_in[9];
    const float* b_hh      = (const float*)d_in[10];
    const float* attn_W    = (const float*)d_in[11];
    const float* attn_b    = (const float*)d_in[12];
    const float* cattn_W   = (const float*)d_in[13];
    const float* cattn_b   = (const float*)d_in[14];
    const float* gen_W     = (const float*)d_in[15];
    const float* gen_b     = (const float*)d_in[16];
    const float* sig_b     = (const float*)d_in[17];
    const float* out_W     = (const float*)d_in[18];
    const float* out_b     = (const float*)d_in[19];

    float* out = (float*)d_out;
    float* out_log = out;                               // B x WIDE
    float* out_h   = out + (size_t)BB * WIDE;           // B x H
    float* out_c   = out_h + (size_t)BB * HH;           // B x H

    // Workspace carve-up (floats)
    float* ws = (float*)d_ws;
    float* prob    = ws;                    ws += (size_t)BB * WIDE;   // 6.41M
    float* x       = ws;                    ws += (size_t)BB * EE;
    float* gates   = ws;                    ws += (size_t)BB * 4 * HH;
    float* dec     = ws;                    ws += (size_t)BB * HH;
    float* c_new   = ws;                    ws += (size_t)BB * HH;
    float* d1      = ws;                    ws += (size_t)BB * HH;
    float* d2      = ws;                    ws += (size_t)BB * HH;
    float* scores1 = ws;                    ws += (size_t)BB * TT;
    float* cscores = ws;                    ws += (size_t)BB * TCC;
    float* ctxv    = ws;                    ws += (size_t)BB * HH;
    float* cctxv   = ws;                    ws += (size_t)BB * HH;
    float* p_gen   = ws;                    ws += BB;

    // 1. embedding
    embed_kernel<<<BB, 256, 0, stream>>>(input_ids, emb, x);

    // 2. LSTM gates = x @ W_ih^T + b_ih ; += h0 @ W_hh^T + b_hh
    gemm_xwT_wmma<<<dim3(4 * HH / 16, BB / 16), 32, 0, stream>>>(
        x, EE, W_ih, EE, b_ih, gates, 4 * HH, EE, 0);
    gemm_xwT_wmma<<<dim3(4 * HH / 16, BB / 16), 32, 0, stream>>>(
        h0, HH, W_hh, HH, b_hh, gates, 4 * HH, HH, 1);

    // 3. LSTM pointwise -> dec (=h_new), c_new (also into output tail)
    lstm_pointwise<<<(BB * HH + 255) / 256, 256, 0, stream>>>(
        gates, c0, dec, c_new, out_h, out_c);

    // 4. attention projections d1 = dec@attn_W^T+attn_b, d2 = dec@cattn_W^T+cattn_b
    gemm_xwT_wmma<<<dim3(HH / 16, BB / 16), 32, 0, stream>>>(
        dec, HH, attn_W, HH, attn_b, d1, HH, HH, 0);
    gemm_xwT_wmma<<<dim3(HH / 16, BB / 16), 32, 0, stream>>>(
        dec, HH, cattn_W, HH, cattn_b, d2, HH, HH, 0);

    // 5. source attention: scores1 -> softmax -> context
    attn_scores_kernel<<<dim3(TT / 8, BB), 256, 0, stream>>>(enc, d1, scores1, TT);
    softmax_rows<<<BB, 256, 0, stream>>>(scores1, TT);
    attn_context_kernel<<<BB, HH, 0, stream>>>(enc, scores1, ctxv, TT);

    // 6. context attention: cscores -> softmax -> context_context
    attn_scores_kernel<<<dim3(TCC / 8, BB), 256, 0, stream>>>(cenc, d2, cscores, TCC);
    softmax_rows<<<BB, 256, 0, stream>>>(cscores, TCC);
    attn_context_kernel<<<BB, HH, 0, stream>>>(cenc, cscores, cctxv, TCC);

    // 7. p_gen
    pgen_kernel<<<BB, 256, 0, stream>>>(ctxv, cctxv, dec, x, gen_W, gen_b, sig_b,
                                        ctx_in, p_gen);

    // 8. vocab logits into prob rows (width WIDE), then p_gen-scaled softmax
    gemm_xwT_wmma<<<dim3(VTT / 16, BB / 16), 32, 0, stream>>>(
        dec, HH, out_W, HH, out_b, prob, WIDE, HH, 0);
    vocab_softmax_scale<<<BB, 256, 0, stream>>>(prob, p_gen);

    // 9. copy distribution scatter-add
    scatter_copy_kernel<<<(BB * TCC + 255) / 256, 256, 0, stream>>>(
        ctx_in, cscores, p_gen, prob);

    // 10. log-clip into output
    int n = BB * WIDE;
    log_kernel<<<(n + 255) / 256, 256, 0, stream>>>(prob, out_log, n);
}